// LeastSquareEqualizer_20100446945876
// MI455X (gfx1250) — compile-verified
//
#include <hip/hip_runtime.h>

typedef float v2f __attribute__((ext_vector_type(2)));
typedef float v4f __attribute__((ext_vector_type(4)));
typedef int   v2i __attribute__((ext_vector_type(2)));

#define NUM_EFF    3696
#define NUM_DATA   44352
#define FFT_N      4096
#define NUM_OFDM   14
#define NUM_XHAT   11354112   // B(128) * tx(2) * s(1) * NUM_DATA  (complex elements)
// d_out: x_hat = NUM_XHAT*2 floats (interleaved complex), then no_eff = NUM_XHAT floats.

__global__ __launch_bounds__(256)
void LeastSquareEqualizer_kernel(const float* __restrict__ y,          // complex64 [B,2,1,14,4096]
                                 const float* __restrict__ h,          // complex64 [B,2,1,2,1,14,3696]
                                 const float* __restrict__ no_p,       // scalar
                                 const int*   __restrict__ eff_sc,     // [3696]
                                 const int*   __restrict__ ind_des,    // [2]
                                 const int*   __restrict__ stream_ind, // [2]
                                 const int*   __restrict__ data_ind,   // [2,1,44352]
                                 float*       __restrict__ out)
{
    __shared__ int s_ind[1024];

    const int q    = blockIdx.x * 256 + threadIdx.x;
    const int idx0 = q * 4;                    // flat output index: (b*2+tx)*44352 + d
    const int d    = idx0 % NUM_DATA;          // multiple of 4; quad shares (b,tx)
    const int t    = idx0 / NUM_DATA;          // b*2 + tx
    const int tx   = t & 1;
    const int b    = t >> 1;

    // --- CDNA5 async copy: stage this thread's four data_ind entries into LDS
    //     (tracked by ASYNCcnt), overlapped with the metadata loads below.
    const int* gsrc = data_ind + tx * NUM_DATA + d;
    unsigned lds_off = (unsigned)(size_t)(&s_ind[threadIdx.x * 4]);
    asm volatile("global_load_async_to_lds_b128 %0, %1, off"
                 :: "v"(lds_off), "v"(gsrc) : "memory");

    // metadata: load both 2-entry tables as single b64s (independent, no chained gather)
    const v2i st2 = *(const v2i*)stream_ind;
    const v2i de2 = *(const v2i*)ind_des;
    const float no = no_p[0];
    const int si   = tx ? st2.y : st2.x;       // rx index for y (NUM_RX_ANT==1)
    const int des  = (si == 0) ? de2.x : de2.y;
    const int rx_h = des >> 1;
    const int tx_h = des & 1;

    const int ybase = (b * 2 + si) * (NUM_OFDM * FFT_N);
    const int hbase = ((b * 2 + rx_h) * 2 + tx_h) * (NUM_OFDM * NUM_EFF);
    const v2f* yp = (const v2f*)y;
    const v2f* hp = (const v2f*)h;

    asm volatile("s_wait_asynccnt 0" ::: "memory");
    // each thread reads only its own LDS slots -> no barrier needed

    v4f xo0, xo1;
    float xr[4], xi[4];
#pragma unroll
    for (int j = 0; j < 4; ++j) {
        const int flat = s_ind[threadIdx.x * 4 + j];
        const int of   = flat / NUM_EFF;
        const int e    = flat - of * NUM_EFF;
        const int sc   = eff_sc[e];

        // read-once streams: NT so the index tables stay L2-resident
        v2f yv = __builtin_nontemporal_load(yp + ybase + of * FFT_N  + sc);
        v2f hv = __builtin_nontemporal_load(hp + hbase + of * NUM_EFF + e);

        // x = y * conj(h) / |h|^2 ; v_rcp_f32 + one Newton step (~0.5 ulp)
        float den = hv.x * hv.x + hv.y * hv.y;
        float r = __builtin_amdgcn_rcpf(den);
        r = r * (2.0f - den * r);
        xr[j] = (yv.x * hv.x + yv.y * hv.y) * r;
        xi[j] = (yv.y * hv.x - yv.x * hv.y) * r;
    }
    xo0.x = xr[0]; xo0.y = xi[0]; xo0.z = xr[1]; xo0.w = xi[1];
    xo1.x = xr[2]; xo1.y = xi[2]; xo1.z = xr[3]; xo1.w = xi[3];

    // 32B-aligned NT stores: four complex outputs
    float* xbase = out + (size_t)idx0 * 2;
    __builtin_nontemporal_store(xo0, (v4f*)(xbase));
    __builtin_nontemporal_store(xo1, (v4f*)(xbase + 4));

    // no_eff = no broadcast; full b128 NT store
    v4f nv; nv.x = no; nv.y = no; nv.z = no; nv.w = no;
    __builtin_nontemporal_store(nv, (v4f*)(out + (size_t)NUM_XHAT * 2 + idx0));
}

extern "C" void kernel_launch(void* const* d_in, const int* in_sizes, int n_in,
                              void* d_out, int out_size, void* d_ws, size_t ws_size,
                              hipStream_t stream) {
    const float* y          = (const float*)d_in[0];
    const float* h_hat      = (const float*)d_in[1];
    const float* no         = (const float*)d_in[2];
    const int*   eff_sc     = (const int*)d_in[3];
    const int*   ind_des    = (const int*)d_in[4];
    const int*   stream_ind = (const int*)d_in[5];
    const int*   data_ind   = (const int*)d_in[6];

    // 11,354,112 complex outputs, 4 per thread -> exactly 11088 blocks of 256
    dim3 grid(11088), block(256);
    hipLaunchKernelGGL(LeastSquareEqualizer_kernel, grid, block, 0, stream,
                       y, h_hat, no, eff_sc, ind_des, stream_ind, data_ind,
                       (float*)d_out);
}